// RM_SEW_37503654428915
// MI455X (gfx1250) — compile-verified
//
#include <hip/hip_runtime.h>
#include <hip/hip_bf16.h>
#include <math.h>

typedef float v2f __attribute__((ext_vector_type(2)));
typedef float v8f __attribute__((ext_vector_type(8)));

#define B  16
#define F  8
#define C  256
#define HW 1024                 // 32*32
#define TILES (B * F * C)       // 32768
#define KSEL 205                // ceil(0.8 * 256)

// ---------------------------------------------------------------------------
// Pass 1: per-(b,f,c) sum & max over the 1024 contiguous h*w elements.
// One 256-thread block per tile; each thread does one float4 (b128) load.
// ---------------------------------------------------------------------------
__global__ __launch_bounds__(256) void pool_kernel(const float* __restrict__ x,
                                                   float* __restrict__ tsum,
                                                   float* __restrict__ tmax) {
    __shared__ float s_s[256];
    __shared__ float s_m[256];
    const int t = blockIdx.x;
    const float4 v = ((const float4*)(x + (size_t)t * HW))[threadIdx.x];
    s_s[threadIdx.x] = (v.x + v.y) + (v.z + v.w);
    s_m[threadIdx.x] = fmaxf(fmaxf(v.x, v.y), fmaxf(v.z, v.w));
    __syncthreads();
    for (int off = 128; off > 0; off >>= 1) {
        if (threadIdx.x < off) {
            s_s[threadIdx.x] += s_s[threadIdx.x + off];
            s_m[threadIdx.x] = fmaxf(s_m[threadIdx.x], s_m[threadIdx.x + off]);
        }
        __syncthreads();
    }
    if (threadIdx.x == 0) { tsum[t] = s_s[0]; tmax[t] = s_m[0]; }
}

// ---------------------------------------------------------------------------
// [16,256] @ [256,256]^T GEMM on V_WMMA_F32_16X16X4_F32 (exact f32).
//   out[m,n] = sum_k W[n,k] * A[m,k]     (W row-major [256,256])
// 512 threads = 16 waves; wave w owns output cols n = 16w..16w+15.
// f32 16x4 A layout: VGPR v holds K = v + 2*(lane>=16), row M = lane&15.
// f32 4x16 B layout: VGPR v holds K = v + 2*(lane>=16), col N = lane&15.
// C/D layout: VGPR r holds M = r + 8*(lane>=16), N = lane&15.
// ---------------------------------------------------------------------------
template <bool RELU, bool ACCUM>
__device__ inline void gemm256(const float* __restrict__ sA,   // LDS [16,256]
                               const float* __restrict__ W,    // global [256,256]
                               float* __restrict__ sOut) {     // LDS [16,256]
    const int lane = threadIdx.x & 31;
    const int wv   = threadIdx.x >> 5;          // 0..15
    const int m    = lane & 15;
    const int kk   = (lane >> 4) << 1;          // 0 or 2
    const int n    = (wv << 4) + (lane & 15);   // output column

    const float* __restrict__ arow = sA + m * 256;
    const float* __restrict__ brow = W  + n * 256;

    v8f acc = {0.f, 0.f, 0.f, 0.f, 0.f, 0.f, 0.f, 0.f};
    for (int k0 = 0; k0 < 256; k0 += 4) {
        v2f a, b;
        a.x = arow[k0 + kk];
        a.y = arow[k0 + kk + 1];
        b.x = brow[k0 + kk];
        b.y = brow[k0 + kk + 1];
        acc = __builtin_amdgcn_wmma_f32_16x16x4_f32(
            /*neg_a=*/false, a, /*neg_b=*/false, b,
            /*c_mod=*/(short)0, acc, /*reuse_a=*/false, /*reuse_b=*/false);
    }

    const int mbase = (lane >> 4) << 3;         // 0 or 8
#pragma unroll
    for (int r = 0; r < 8; ++r) {
        float v = acc[r];
        if (RELU) v = fmaxf(v, 0.f);
        const int idx = (mbase + r) * 256 + n;
        if (ACCUM) sOut[idx] += v; else sOut[idx] = v;
    }
}

// ---------------------------------------------------------------------------
// Single-workgroup middle stage: time pools, ta MLP+sigmoid, channel pools,
// four WMMA GEMMs, sigmoid, per-row 205th-largest threshold, scale table.
// ---------------------------------------------------------------------------
__global__ __launch_bounds__(512) void attn_kernel(const float* __restrict__ tsum,
                                                   const float* __restrict__ tmax,
                                                   const float* __restrict__ ta_w1,
                                                   const float* __restrict__ ta_w2,
                                                   const float* __restrict__ ca_w1,
                                                   const float* __restrict__ ca_w2,
                                                   float* __restrict__ scale) {
    __shared__ float s_tavg[B * F], s_tmax[B * F];
    __shared__ float s_ha[B * F], s_hm[B * F], s_ta[B * F];
    __shared__ float s_thr[B];
    __shared__ float s_A[B * C];    // 16 KB: channel pool / GEMM input
    __shared__ float s_H[B * C];    // 16 KB: relu(hidden)
    __shared__ float s_O[B * C];    // 16 KB: MLP out -> ca
    const int tid = threadIdx.x;

    // ---- time pools: mean/max over c (per h*w tile sums/maxes) ----
    if (tid < B * F) {
        const float* ps = tsum + tid * C;
        const float* pm = tmax + tid * C;
        float s = 0.f, m = -INFINITY;
        for (int c = 0; c < C; ++c) { s += ps[c]; m = fmaxf(m, pm[c]); }
        s_tavg[tid] = s * (1.f / (float)(C * HW));
        s_tmax[tid] = m;
    }
    __syncthreads();

    // ---- time-attention 8x8 MLP (shared weights, avg+max paths) ----
    if (tid < B * F) {
        const int b = tid >> 3, h = tid & 7;
        float sa = 0.f, sm = 0.f;
        for (int n = 0; n < F; ++n) {
            const float w = ta_w1[h * F + n];
            sa += w * s_tavg[b * F + n];
            sm += w * s_tmax[b * F + n];
        }
        s_ha[tid] = fmaxf(sa, 0.f);
        s_hm[tid] = fmaxf(sm, 0.f);
    }
    __syncthreads();
    if (tid < B * F) {
        const int b = tid >> 3, n = tid & 7;
        float o = 0.f;
        for (int h = 0; h < F; ++h)
            o += ta_w2[n * F + h] * (s_ha[b * F + h] + s_hm[b * F + h]);
        s_ta[tid] = 1.f / (1.f + expf(-o));   // WTA on ta: k=ceil(.9*8)=8 -> identity
    }
    __syncthreads();

    // ---- channel avg pool of ta*x:  c_avg[b,c] = sum_f ta*tsum / (F*HW) ----
    for (int idx = tid; idx < B * C; idx += 512) {
        const int b = idx >> 8, c = idx & 255;
        float s = 0.f;
        for (int f = 0; f < F; ++f)
            s += s_ta[b * F + f] * tsum[(b * F + f) * C + c];
        s_A[idx] = s * (1.f / (float)(F * HW));
    }
    __syncthreads();
    gemm256<true, false>(s_A, ca_w1, s_H);    // hidden = relu(c_avg @ W1^T)
    __syncthreads();
    gemm256<false, false>(s_H, ca_w2, s_O);   // out    = hidden @ W2^T
    __syncthreads();

    // ---- channel max pool: ta>0 so max commutes with scaling ----
    for (int idx = tid; idx < B * C; idx += 512) {
        const int b = idx >> 8, c = idx & 255;
        float m = -INFINITY;
        for (int f = 0; f < F; ++f)
            m = fmaxf(m, s_ta[b * F + f] * tmax[(b * F + f) * C + c]);
        s_A[idx] = m;
    }
    __syncthreads();
    gemm256<true, false>(s_A, ca_w1, s_H);
    __syncthreads();
    gemm256<false, true>(s_H, ca_w2, s_O);    // accumulate avg+max paths
    __syncthreads();

    // ---- sigmoid -> ca ----
    for (int idx = tid; idx < B * C; idx += 512)
        s_O[idx] = 1.f / (1.f + expf(-s_O[idx]));
    __syncthreads();

    // ---- per-row 205th-largest (threshold), tie-robust count selection ----
    for (int idx = tid; idx < B * C; idx += 512) {
        const int b = idx >> 8;
        const float v = s_O[idx];
        const float* __restrict__ row = s_O + (b << 8);
        int cg = 0, ce = 0;
        for (int j = 0; j < C; ++j) {
            const float u = row[j];
            cg += (u > v);
            ce += (u >= v);
        }
        if (cg <= KSEL - 1 && ce >= KSEL) s_thr[b] = v;  // all writers agree
    }
    __syncthreads();

    // ---- scale[b,f,c] = ta[b,f] * ca[b,c] * (ca >= thr[b]) ----
    for (int idx = tid; idx < TILES; idx += 512) {
        const int b = idx >> 11, f = (idx >> 8) & 7, c = idx & 255;
        const float ca = s_O[(b << 8) + c];
        const float sc = (ca >= s_thr[b]) ? ca : 0.f;
        scale[idx] = s_ta[b * F + f] * sc;
    }
}

// ---------------------------------------------------------------------------
// Pass 2: out = x * scale[b,f,c].  x is L2-resident (128 MB < 192 MB L2)
// after pass 1, so this is ~write-bandwidth bound.
// ---------------------------------------------------------------------------
__global__ __launch_bounds__(256) void apply_kernel(const float* __restrict__ x,
                                                    const float* __restrict__ scale,
                                                    float* __restrict__ out) {
    const int t = blockIdx.x;
    const float s = scale[t];
    float4 v = ((const float4*)(x + (size_t)t * HW))[threadIdx.x];
    v.x *= s; v.y *= s; v.z *= s; v.w *= s;
    ((float4*)(out + (size_t)t * HW))[threadIdx.x] = v;
}

extern "C" void kernel_launch(void* const* d_in, const int* in_sizes, int n_in,
                              void* d_out, int out_size, void* d_ws, size_t ws_size,
                              hipStream_t stream) {
    const float* x     = (const float*)d_in[0];
    const float* ta_w1 = (const float*)d_in[1];
    const float* ta_w2 = (const float*)d_in[2];
    const float* ca_w1 = (const float*)d_in[3];
    const float* ca_w2 = (const float*)d_in[4];
    float* out = (float*)d_out;

    float* tsum  = (float*)d_ws;        // 32768 f32
    float* tmax  = tsum + TILES;        // 32768 f32
    float* scale = tmax + TILES;        // 32768 f32   (384 KB total ws)

    pool_kernel<<<TILES, 256, 0, stream>>>(x, tsum, tmax);
    attn_kernel<<<1, 512, 0, stream>>>(tsum, tmax, ta_w1, ta_w2, ca_w1, ca_w2, scale);
    apply_kernel<<<TILES, 256, 0, stream>>>(x, scale, out);
}